// BertKG_63015760167157
// MI455X (gfx1250) — compile-verified
//
#include <hip/hip_runtime.h>

typedef __attribute__((ext_vector_type(2))) float v2f;
typedef __attribute__((ext_vector_type(8))) float v8f;

#define N_ITEMS 30000
#define N_USERS 20000
#define N_NODES 50000
#define N_EDGES 512000
#define D0 768
#define D1 128
#define D2 64
#define OUT_D (D0 + D1 + D2)   // 960

// ---------------------------------------------------------------- utilities

__global__ void zero_f32x4(float* __restrict__ p, long long n4) {
    long long i = (long long)blockIdx.x * blockDim.x + threadIdx.x;
    if (i < n4) ((float4*)p)[i] = make_float4(0.f, 0.f, 0.f, 0.f);
}

// copy ego0 = concat(item, user) into out[:, 0:768] (row stride OUT_D)
__global__ void copy_ego0(const float* __restrict__ item,
                          const float* __restrict__ user,
                          float* __restrict__ out) {
    long long t = (long long)blockIdx.x * blockDim.x + threadIdx.x;
    const long long total = (long long)N_NODES * (D0 / 4);
    if (t >= total) return;
    int row = (int)(t / (D0 / 4));
    int c   = (int)(t % (D0 / 4)) * 4;
    const float* s = (row < N_ITEMS) ? (item + (size_t)row * D0)
                                     : (user + (size_t)(row - N_ITEMS) * D0);
    float4 x = *(const float4*)(s + c);
    *(float4*)(out + (size_t)row * OUT_D + c) = x;
}

// ------------------------------------------------------- sparse A @ X (scatter)

template <int D, bool CONCAT>
__global__ void spmm_scatter(const float* __restrict__ vals,
                             const int*   __restrict__ rows,
                             const int*   __restrict__ cols,
                             const float* __restrict__ srcA,   // item (CONCAT) or dense src
                             const float* __restrict__ srcB,   // user (CONCAT) or unused
                             float* __restrict__ side) {
    constexpr int CPE = D / 4;  // float4 chunks per edge
    long long t = (long long)blockIdx.x * blockDim.x + threadIdx.x;
    if (t >= (long long)N_EDGES * CPE) return;
    int e = (int)(t / CPE);
    int c = (int)(t % CPE) * 4;
    float v  = vals[e];
    int col  = cols[e];
    int row  = rows[e];
    const float* s = CONCAT
        ? ((col < N_ITEMS) ? srcA + (size_t)col * D
                           : srcB + (size_t)(col - N_ITEMS) * D)
        : srcA + (size_t)col * D;
    float4 x = *(const float4*)(s + c);
    float* d = side + (size_t)row * D + c;
    __hip_atomic_fetch_add(d + 0, v * x.x, __ATOMIC_RELAXED, __HIP_MEMORY_SCOPE_AGENT);
    __hip_atomic_fetch_add(d + 1, v * x.y, __ATOMIC_RELAXED, __HIP_MEMORY_SCOPE_AGENT);
    __hip_atomic_fetch_add(d + 2, v * x.z, __ATOMIC_RELAXED, __HIP_MEMORY_SCOPE_AGENT);
    __hip_atomic_fetch_add(d + 3, v * x.w, __ATOMIC_RELAXED, __HIP_MEMORY_SCOPE_AGENT);
}

// ------------------------------------------------ fused NGCF dense layer (WMMA)
// out = leaky_relu( (side+ego) @ W1 + b1 + (side*ego) @ W2 + b2 )
// One wave (32 lanes) per 16x16 output tile; threadIdx.y selects the N-tile.
// A layout (f32 16x4): lane L holds row M=L%16, K = 2*(L/16)+{0,1}.
// B layout (f32 4x16): lane L holds col N=L%16, K = 2*(L/16)+{0,1}.
// C/D layout: VGPR v, lane L -> M = v + 8*(L/16), N = L%16.

template <int DK, int DN, bool CONCAT>
__global__ void ngcf_dense(const float* __restrict__ side,
                           const float* __restrict__ egoA,   // item (CONCAT) or ego
                           const float* __restrict__ egoB,   // user (CONCAT) or unused
                           const float* __restrict__ W1, const float* __restrict__ b1,
                           const float* __restrict__ W2, const float* __restrict__ b2,
                           float* __restrict__ out) {
    const int tile_m = blockIdx.x;          // N_NODES/16 tiles (exact: 3125)
    const int tile_n = threadIdx.y;         // DN/16 tiles
    const int lane   = threadIdx.x;         // 0..31
    const int m      = lane & 15;
    const int kq     = (lane >> 4) * 2;     // 0 or 2
    const int row    = tile_m * 16 + m;
    const int gn     = tile_n * 16 + (lane & 15);

    const float* erow = CONCAT
        ? ((row < N_ITEMS) ? egoA + (size_t)row * DK
                           : egoB + (size_t)(row - N_ITEMS) * DK)
        : egoA + (size_t)row * DK;
    const float* srow = side + (size_t)row * DK;

    v8f c = {};
    for (int k = 0; k < DK; k += 4) {
        const int k0 = k + kq;
        v2f sv = *(const v2f*)(srow + k0);
        v2f ev = *(const v2f*)(erow + k0);
        v2f a1 = sv + ev;          // (A + I) @ ego term
        v2f a2 = sv * ev;          // Hadamard term
        v2f bw1, bw2;
        bw1.x = W1[(size_t)k0 * DN + gn];
        bw1.y = W1[(size_t)(k0 + 1) * DN + gn];
        bw2.x = W2[(size_t)k0 * DN + gn];
        bw2.y = W2[(size_t)(k0 + 1) * DN + gn];
        c = __builtin_amdgcn_wmma_f32_16x16x4_f32(false, a1, false, bw1,
                                                  (short)0, c, false, false);
        c = __builtin_amdgcn_wmma_f32_16x16x4_f32(false, a2, false, bw2,
                                                  (short)0, c, false, false);
    }

    const float bias  = b1[gn] + b2[gn];
    const int   mbase = tile_m * 16 + (lane >> 4) * 8;
    #pragma unroll
    for (int v = 0; v < 8; ++v) {
        float x = c[v] + bias;
        x = (x >= 0.f) ? x : 0.01f * x;     // leaky_relu, slope 0.01
        out[(size_t)(mbase + v) * DN + gn] = x;
    }
}

// ------------------------------------------------------------- row L2 normalize

template <int D>
__global__ void row_normalize(const float* __restrict__ src,
                              float* __restrict__ dst,
                              int dst_stride, int dst_off) {
    const int row  = blockIdx.x;
    const int lane = threadIdx.x;           // 32 lanes
    const float* s = src + (size_t)row * D;
    float vals[D / 32];
    float ss = 0.f;
    #pragma unroll
    for (int i = 0; i < D / 32; ++i) {
        float x = s[lane + 32 * i];
        vals[i] = x;
        ss += x * x;
    }
    #pragma unroll
    for (int off = 16; off >= 1; off >>= 1)
        ss += __shfl_xor(ss, off, 32);
    float inv = 1.f / fmaxf(sqrtf(ss), 1e-12f);
    float* d = dst + (size_t)row * dst_stride + dst_off;
    #pragma unroll
    for (int i = 0; i < D / 32; ++i)
        d[lane + 32 * i] = vals[i] * inv;
}

// ---------------------------------------------------------------------- launch

extern "C" void kernel_launch(void* const* d_in, const int* in_sizes, int n_in,
                              void* d_out, int out_size, void* d_ws, size_t ws_size,
                              hipStream_t stream) {
    const float* item  = (const float*)d_in[0];
    const float* user  = (const float*)d_in[1];
    const float* W1_0  = (const float*)d_in[2];
    const float* b1_0  = (const float*)d_in[3];
    const float* W2_0  = (const float*)d_in[4];
    const float* b2_0  = (const float*)d_in[5];
    const float* W1_1  = (const float*)d_in[6];
    const float* b1_1  = (const float*)d_in[7];
    const float* W2_1  = (const float*)d_in[8];
    const float* b2_1  = (const float*)d_in[9];
    const float* evals = (const float*)d_in[10];
    const int*   erows = (const int*)d_in[11];
    const int*   ecols = (const int*)d_in[12];
    float* out = (float*)d_out;

    float* side0 = (float*)d_ws;                       // 50000*768
    float* ego1  = side0 + (size_t)N_NODES * D0;       // 50000*128
    float* side1 = ego1  + (size_t)N_NODES * D1;       // 50000*128
    float* ego2  = side1 + (size_t)N_NODES * D1;       // 50000*64

    const int TB = 256;

    // zero side0; emit ego0 slice of output
    {
        long long n4 = (long long)N_NODES * D0 / 4;
        zero_f32x4<<<(unsigned)((n4 + TB - 1) / TB), TB, 0, stream>>>(side0, n4);
        long long nc = (long long)N_NODES * (D0 / 4);
        copy_ego0<<<(unsigned)((nc + TB - 1) / TB), TB, 0, stream>>>(item, user, out);
    }

    // layer 0: side0 = A @ ego0 (gather + atomic scatter)
    {
        long long nt = (long long)N_EDGES * (D0 / 4);
        spmm_scatter<D0, true><<<(unsigned)((nt + TB - 1) / TB), TB, 0, stream>>>(
            evals, erows, ecols, item, user, side0);
    }
    // layer 0 dense: ego1 = leaky_relu((side0+ego0)@W1_0 + (side0*ego0)@W2_0 + b)
    ngcf_dense<D0, D1, true><<<dim3(N_NODES / 16), dim3(32, D1 / 16), 0, stream>>>(
        side0, item, user, W1_0, b1_0, W2_0, b2_0, ego1);

    // layer 1: side1 = A @ ego1
    {
        long long n4 = (long long)N_NODES * D1 / 4;
        zero_f32x4<<<(unsigned)((n4 + TB - 1) / TB), TB, 0, stream>>>(side1, n4);
        long long nt = (long long)N_EDGES * (D1 / 4);
        spmm_scatter<D1, false><<<(unsigned)((nt + TB - 1) / TB), TB, 0, stream>>>(
            evals, erows, ecols, ego1, nullptr, side1);
    }
    // layer 1 dense: ego2
    ngcf_dense<D1, D2, false><<<dim3(N_NODES / 16), dim3(32, D2 / 16), 0, stream>>>(
        side1, ego1, nullptr, W1_1, b1_1, W2_1, b2_1, ego2);

    // normalized slices into output (note: next-layer input stays UN-normalized,
    // matching the reference, which appends ego/norm but carries raw ego forward)
    row_normalize<D1><<<N_NODES, 32, 0, stream>>>(ego1, out, OUT_D, D0);
    row_normalize<D2><<<N_NODES, 32, 0, stream>>>(ego2, out, OUT_D, D0 + D1);
}